// SelfAttention_1958505087074
// MI455X (gfx1250) — compile-verified
//
#include <hip/hip_runtime.h>
#include <math.h>

#define MD    1024
#define NH    16
#define HD    64
#define BATCH 4
#define SEQ   2048

typedef __attribute__((ext_vector_type(2))) float v2f;
typedef __attribute__((ext_vector_type(8))) float v8f;

static __device__ __forceinline__ v8f wmma_f32(v2f a, v2f b, v8f c) {
  // D = A(16x4, f32) * B(4x16, f32) + C(16x16, f32)
  return __builtin_amdgcn_wmma_f32_16x16x4_f32(
      /*neg_a=*/false, a, /*neg_b=*/false, b,
      /*c_mod=*/(short)0, c, /*reuse_a=*/false, /*reuse_b=*/false);
}

// ---------------------------------------------------------------------------
// C[m, n] = sum_k A[m,k] * W[n,k]  (+ bias[n])      "NT" layout GEMM
// A: [M=8192, 1024] row-major.  W: [1024, 1024] row-major (out, in).
// headsplit=1: write C to [B, H, SEQ, D] (for Q/K/V); else [M, 1024] (+bias).
// Block tile 128x64, 8 waves in 4(M) x 2(N); each wave computes a 32x32
// output patch as 2x2 WMMA accumulators -> 4 LDS frag loads per 4 WMMAs.
// ---------------------------------------------------------------------------
__global__ __launch_bounds__(256)
void gemm_nt_128x64(const float* __restrict__ A, const float* __restrict__ W,
                    const float* __restrict__ bias, float* __restrict__ C,
                    int headsplit)
{
  __shared__ float As[128][18];  // stride 18: conflict-free 16-row frag reads
  __shared__ float Ws[64][18];

  const int m0   = blockIdx.x * 128;
  const int n0   = blockIdx.y * 64;
  const int tid  = threadIdx.x;
  const int w    = tid >> 5;
  const int lane = tid & 31;
  const int ll   = lane & 15;
  const int hh   = lane >> 4;        // half-wave select
  const int mq   = (w >> 1) * 32;    // wave M offset within block: 0/32/64/96
  const int nq   = (w & 1) * 32;     // wave N offset within block: 0/32

  const v8f z = {0.f,0.f,0.f,0.f,0.f,0.f,0.f,0.f};
  v8f acc00 = z, acc01 = z, acc10 = z, acc11 = z;  // acc[mi][ni]

  const int lrA0 = tid >> 2;          // A loader rows: lrA0, lrA0+64
  const int lcA  = (tid & 3) * 4;     // col 0/4/8/12
  const int lrW  = tid >> 2;          // W loader row 0..63

  for (int k0 = 0; k0 < MD; k0 += 16) {
    const float4 a0 = *(const float4*)&A[(size_t)(m0 + lrA0     ) * MD + k0 + lcA];
    const float4 a1 = *(const float4*)&A[(size_t)(m0 + lrA0 + 64) * MD + k0 + lcA];
    const float4 wv = *(const float4*)&W[(size_t)(n0 + lrW      ) * MD + k0 + lcA];
    As[lrA0   ][lcA+0] = a0.x; As[lrA0   ][lcA+1] = a0.y;
    As[lrA0   ][lcA+2] = a0.z; As[lrA0   ][lcA+3] = a0.w;
    As[lrA0+64][lcA+0] = a1.x; As[lrA0+64][lcA+1] = a1.y;
    As[lrA0+64][lcA+2] = a1.z; As[lrA0+64][lcA+3] = a1.w;
    Ws[lrW][lcA+0] = wv.x; Ws[lrW][lcA+1] = wv.y;
    Ws[lrW][lcA+2] = wv.z; Ws[lrW][lcA+3] = wv.w;
    __syncthreads();

    if (k0 + 16 < MD) {   // global_prefetch_b8 for the next K tile
      __builtin_prefetch(&A[(size_t)(m0 + lrA0     ) * MD + k0 + 16 + lcA], 0, 1);
      __builtin_prefetch(&A[(size_t)(m0 + lrA0 + 64) * MD + k0 + 16 + lcA], 0, 1);
      __builtin_prefetch(&W[(size_t)(n0 + lrW      ) * MD + k0 + 16 + lcA], 0, 1);
    }

#pragma unroll
    for (int kk = 0; kk < 16; kk += 4) {
      const int col = kk + 2 * hh;   // lane<16 -> {K,K+1}; lane>=16 -> {K+2,K+3}
      const v2f a0f = *(const v2f*)&As[mq      + ll][col];
      const v2f a1f = *(const v2f*)&As[mq + 16 + ll][col];
      const v2f b0f = *(const v2f*)&Ws[nq      + ll][col];
      const v2f b1f = *(const v2f*)&Ws[nq + 16 + ll][col];
      acc00 = wmma_f32(a0f, b0f, acc00);
      acc01 = wmma_f32(a0f, b1f, acc01);
      acc10 = wmma_f32(a1f, b0f, acc10);
      acc11 = wmma_f32(a1f, b1f, acc11);
    }
    __syncthreads();
  }

  // Epilogue: C-fragment layout -> vgpr j: lanes0-15 M=j, lanes16-31 M=j+8.
#pragma unroll
  for (int mi = 0; mi < 2; ++mi) {
#pragma unroll
    for (int ni = 0; ni < 2; ++ni) {
      const v8f acc = (mi == 0) ? (ni == 0 ? acc00 : acc01)
                                : (ni == 0 ? acc10 : acc11);
#pragma unroll
      for (int j = 0; j < 8; ++j) {
        const int gm = m0 + mq + mi * 16 + j + hh * 8;
        const int gn = n0 + nq + ni * 16 + ll;
        float c = acc[j];
        if (!headsplit) {
          c += bias[gn];
          C[(size_t)gm * MD + gn] = c;
        } else {
          const int b = gm / SEQ, n = gm % SEQ;
          const int h = gn >> 6, d = gn & 63;
          C[(((size_t)b * NH + h) * SEQ + n) * HD + d] = c;
        }
      }
    }
  }
}

// ---------------------------------------------------------------------------
// Causal flash attention. Q/K/V: [B*H, SEQ, 64] f32. One block = 4 waves,
// each wave owns a 16-row query tile (64 rows/block). Key/value tiles of 16
// rows are LDS-staged and shared by all 4 waves; V is stored transposed so
// P*V B-fragments are single v2f LDS loads. Online softmax, f32 WMMA.
// Output written to [B, SEQ, 1024] for the final projection.
// ---------------------------------------------------------------------------
__global__ __launch_bounds__(128)
void flash_attn_causal(const float* __restrict__ Q, const float* __restrict__ K,
                       const float* __restrict__ V, float* __restrict__ O)
{
  __shared__ float Ks[16][66];      // 16 keys x 64 dims (padded)
  __shared__ float Vt[64][18];      // transposed: Vt[d][key] (padded)
  __shared__ float Ps[4][16][18];   // per-wave P tile; 18 -> 8B-aligned rows

  const int bh   = blockIdx.x;      // b*NH + h
  const int qblk = blockIdx.y;      // 0..SEQ/64-1
  const int tid  = threadIdx.x;
  const int w    = tid >> 5;
  const int lane = tid & 31;
  const int ll   = lane & 15;
  const int hh   = lane >> 4;
  const int qbase = qblk * 64 + w * 16;
  const float scale = 0.125f;       // 1/sqrt(64)

  const float* Qh = Q + (size_t)bh * SEQ * HD;
  const float* Kh = K + (size_t)bh * SEQ * HD;
  const float* Vh = V + (size_t)bh * SEQ * HD;

  // Preload the wave's 16x64 Q tile as 16 pre-scaled A-fragments.
  v2f qf[16];
#pragma unroll
  for (int t = 0; t < 16; ++t) {
    v2f qv = *(const v2f*)&Qh[(size_t)(qbase + ll) * HD + t * 4 + 2 * hh];
    qf[t] = qv * scale;
  }

  const v8f z = {0.f,0.f,0.f,0.f,0.f,0.f,0.f,0.f};
  v8f oacc[4] = {z, z, z, z};       // O tile: 16 rows x 64 dims
  float mrow[8], lrow[8];
#pragma unroll
  for (int j = 0; j < 8; ++j) { mrow[j] = -INFINITY; lrow[j] = 0.f; }

  const int nkt = qblk * 4 + 4;     // key tiles needed by this block (causal)
  for (int kt = 0; kt < nkt; ++kt) {
    const int kbase = kt * 16;

    // Cooperative K/V tile load: 128 threads x 2 float4 per tile.
#pragma unroll
    for (int part = 0; part < 2; ++part) {
      const int lin = tid + part * 128;          // 0..255
      const int row = lin >> 4, c4 = (lin & 15) * 4;
      const float4 kd = *(const float4*)&Kh[(size_t)(kbase + row) * HD + c4];
      const float4 vd = *(const float4*)&Vh[(size_t)(kbase + row) * HD + c4];
      Ks[row][c4+0] = kd.x; Ks[row][c4+1] = kd.y;
      Ks[row][c4+2] = kd.z; Ks[row][c4+3] = kd.w;
      Vt[c4+0][row] = vd.x; Vt[c4+1][row] = vd.y;   // transpose on store
      Vt[c4+2][row] = vd.z; Vt[c4+3][row] = vd.w;
    }
    __syncthreads();

    if (kbase <= qbase + 15) {      // wave-uniform: EXEC all-1s inside
      // S = (Q*scale) . K^T  : 16 chained f32 WMMAs over d=64
      v8f s = z;
#pragma unroll
      for (int t = 0; t < 16; ++t) {
        const v2f b = *(const v2f*)&Ks[ll][t * 4 + 2 * hh];
        s = wmma_f32(qf[t], b, s);
      }

      // Online softmax per row (rows live in 16-lane halves of the fragment).
#pragma unroll
      for (int j = 0; j < 8; ++j) {
        const int rowg = qbase + j + hh * 8;
        const int colg = kbase + ll;
        const float sj = (colg > rowg) ? -INFINITY : s[j];
        float red = sj;
        red = fmaxf(red, __shfl_xor(red, 1, 32));
        red = fmaxf(red, __shfl_xor(red, 2, 32));
        red = fmaxf(red, __shfl_xor(red, 4, 32));
        red = fmaxf(red, __shfl_xor(red, 8, 32));
        const float mnew  = fmaxf(mrow[j], red);
        const float p     = __expf(sj - mnew);
        const float alpha = __expf(mrow[j] - mnew);
        mrow[j] = mnew;
        float psum = p;
        psum += __shfl_xor(psum, 1, 32);
        psum += __shfl_xor(psum, 2, 32);
        psum += __shfl_xor(psum, 4, 32);
        psum += __shfl_xor(psum, 8, 32);
        lrow[j] = lrow[j] * alpha + psum;
#pragma unroll
        for (int dt = 0; dt < 4; ++dt) oacc[dt][j] *= alpha;
        Ps[w][j + hh * 8][ll] = p;   // C-layout -> row-major P in LDS
      }

      // O += P * V : P as A-fragments, V^T rows give v2f B-fragments.
#pragma unroll
      for (int t = 0; t < 4; ++t) {
        const v2f a = *(const v2f*)&Ps[w][ll][t * 4 + 2 * hh];
#pragma unroll
        for (int dt = 0; dt < 4; ++dt) {
          const v2f b = *(const v2f*)&Vt[dt * 16 + ll][t * 4 + 2 * hh];
          oacc[dt] = wmma_f32(a, b, oacc[dt]);
        }
      }
    }
    __syncthreads();
  }

  // Normalize and write to [B, SEQ, MD] at column h*64 + d.
  const int b = bh >> 4;
  const int h = bh & 15;
#pragma unroll
  for (int j = 0; j < 8; ++j) {
    const float inv = 1.0f / lrow[j];
    const int rowg = qbase + j + hh * 8;
#pragma unroll
    for (int dt = 0; dt < 4; ++dt) {
      const int d = dt * 16 + ll;
      O[((size_t)b * SEQ + rowg) * MD + h * HD + d] = oacc[dt][j] * inv;
    }
  }
}

// ---------------------------------------------------------------------------
extern "C" void kernel_launch(void* const* d_in, const int* in_sizes, int n_in,
                              void* d_out, int out_size, void* d_ws, size_t ws_size,
                              hipStream_t stream) {
  (void)in_sizes; (void)n_in; (void)out_size; (void)ws_size;
  const float* x  = (const float*)d_in[0];
  const float* Wq = (const float*)d_in[1];
  const float* Wk = (const float*)d_in[2];
  const float* Wv = (const float*)d_in[3];
  const float* Wo = (const float*)d_in[4];
  const float* bo = (const float*)d_in[5];

  float* ws = (float*)d_ws;
  const size_t SZ = (size_t)BATCH * SEQ * MD;       // 8.39M floats each
  float* qb = ws;
  float* kb = ws + SZ;
  float* vb = ws + 2 * SZ;
  float* ao = ws + 3 * SZ;

  const dim3 ggrid(BATCH * SEQ / 128, MD / 64);     // (64, 16)
  gemm_nt_128x64<<<ggrid, 256, 0, stream>>>(x, Wq, nullptr, qb, 1);
  gemm_nt_128x64<<<ggrid, 256, 0, stream>>>(x, Wk, nullptr, kb, 1);
  gemm_nt_128x64<<<ggrid, 256, 0, stream>>>(x, Wv, nullptr, vb, 1);

  flash_attn_causal<<<dim3(BATCH * NH, SEQ / 64), 128, 0, stream>>>(qb, kb, vb, ao);

  gemm_nt_128x64<<<ggrid, 256, 0, stream>>>(ao, Wo, bo, (float*)d_out, 0);
}